// BootstrapLossG2L_70832600645932
// MI455X (gfx1250) — compile-verified
//
#include <hip/hip_runtime.h>

#define D 256
#define EPS 1e-12f

typedef __attribute__((ext_vector_type(2))) float v2f;
typedef __attribute__((ext_vector_type(8))) float v8f;

__device__ __forceinline__ float wave_sum(float v) {
#pragma unroll
  for (int off = 16; off > 0; off >>= 1)
    v += __shfl_xor(v, off, 32);
  return v;
}

// ---------------------------------------------------------------------------
// Kernel A: one wave32 per graph. Binary-search row range in sorted batch,
// stream h rows (b128 loads), L2-normalize per row (wave shuffle reduction),
// accumulate segment sum in registers, write hsum row. Deterministic, no atomics.
// ---------------------------------------------------------------------------
__global__ __launch_bounds__(256)
void seg_norm_sum_kernel(const float* __restrict__ h1,
                         const float* __restrict__ h2,
                         const int*   __restrict__ b32,
                         float* __restrict__ hsum1,
                         float* __restrict__ hsum2,
                         int N, int G)
{
  const int lane = threadIdx.x & 31;
  const int wave = blockIdx.x * (blockDim.x >> 5) + (threadIdx.x >> 5);
  if (wave >= G) return;
  const int g = wave;

  // Detect whether batch is stored as int64 (reference dtype) or int32.
  const bool is64 = (b32[N - 1] == 0) && (b32[N - 2] != 0);
  auto getb = [&](int i) -> int { return is64 ? b32[2 * i] : b32[i]; };

  // lower_bound(g) and lower_bound(g+1)
  int lo = 0, hiN = N;
  while (lo < hiN) { int mid = (lo + hiN) >> 1; if (getb(mid) < g) lo = mid + 1; else hiN = mid; }
  const int beg = lo;
  hiN = N;
  while (lo < hiN) { int mid = (lo + hiN) >> 1; if (getb(mid) <= g) lo = mid + 1; else hiN = mid; }
  const int end = lo;

  float a1[8] = {0,0,0,0,0,0,0,0};
  float a2[8] = {0,0,0,0,0,0,0,0};

  for (int r = beg; r < end; ++r) {
    const float4* p1 = (const float4*)(h1 + (size_t)r * D);
    const float4* p2 = (const float4*)(h2 + (size_t)r * D);
    float4 x10 = p1[lane], x11 = p1[lane + 32];
    float4 x20 = p2[lane], x21 = p2[lane + 32];

    float s1 = x10.x*x10.x + x10.y*x10.y + x10.z*x10.z + x10.w*x10.w
             + x11.x*x11.x + x11.y*x11.y + x11.z*x11.z + x11.w*x11.w;
    float s2 = x20.x*x20.x + x20.y*x20.y + x20.z*x20.z + x20.w*x20.w
             + x21.x*x21.x + x21.y*x21.y + x21.z*x21.z + x21.w*x21.w;
    s1 = wave_sum(s1);
    s2 = wave_sum(s2);
    const float c1 = 1.0f / fmaxf(sqrtf(s1), EPS);
    const float c2 = 1.0f / fmaxf(sqrtf(s2), EPS);

    a1[0] += x10.x * c1; a1[1] += x10.y * c1; a1[2] += x10.z * c1; a1[3] += x10.w * c1;
    a1[4] += x11.x * c1; a1[5] += x11.y * c1; a1[6] += x11.z * c1; a1[7] += x11.w * c1;
    a2[0] += x20.x * c2; a2[1] += x20.y * c2; a2[2] += x20.z * c2; a2[3] += x20.w * c2;
    a2[4] += x21.x * c2; a2[5] += x21.y * c2; a2[6] += x21.z * c2; a2[7] += x21.w * c2;
  }

  float4* o1 = (float4*)(hsum1 + (size_t)g * D);
  float4* o2 = (float4*)(hsum2 + (size_t)g * D);
  o1[lane]      = make_float4(a1[0], a1[1], a1[2], a1[3]);
  o1[lane + 32] = make_float4(a1[4], a1[5], a1[6], a1[7]);
  o2[lane]      = make_float4(a2[0], a2[1], a2[2], a2[3]);
  o2[lane + 32] = make_float4(a2[4], a2[5], a2[6], a2[7]);
}

// ---------------------------------------------------------------------------
// Kernel B: one wave per 16-row tile. Normalize g rows, then compute the
// row-wise dots via V_WMMA_F32_16X16X4_F32 (C = g_tile * hsum_tile^T over
// 64 k-chunks of 4); take the diagonal of C, wave-reduce, emit one partial.
// A layout (16x4 f32):  VGPR0 = A[m][k0 + 2*half], VGPR1 = A[m][k0 + 2*half + 1]
// B layout (4x16 f32):  VGPR0 = B[k0 + 2*half][n], VGPR1 = B[k0 + 2*half + 1][n]
// with m = n = lane&15, half = lane>>4; B[k][n] = hsum[row0+n][k].
// ---------------------------------------------------------------------------
__global__ __launch_bounds__(256)
void wmma_dot_kernel(const float* __restrict__ g1,
                     const float* __restrict__ g2,
                     const float* __restrict__ hs1,
                     const float* __restrict__ hs2,
                     float* __restrict__ partial,
                     int Gtiles)
{
  const int lane = threadIdx.x & 31;
  const int wave = blockIdx.x * (blockDim.x >> 5) + (threadIdx.x >> 5);
  if (wave >= Gtiles) return;
  const int row0 = wave * 16;
  const int m  = lane & 15;
  const int hv = lane >> 4;

  // Per-row inverse norms of the g tiles; lane keeps the one for its row (m).
  float inv1 = 0.0f, inv2 = 0.0f;
  for (int r = 0; r < 16; ++r) {
    const float* p1 = g1 + (size_t)(row0 + r) * D;
    const float* p2 = g2 + (size_t)(row0 + r) * D;
    float s1 = 0.0f, s2 = 0.0f;
#pragma unroll
    for (int j = 0; j < 8; ++j) {
      float v1 = p1[lane + 32 * j]; s1 += v1 * v1;
      float v2 = p2[lane + 32 * j]; s2 += v2 * v2;
    }
    s1 = wave_sum(s1);
    s2 = wave_sum(s2);
    const float i1 = 1.0f / fmaxf(sqrtf(s1), EPS);
    const float i2 = 1.0f / fmaxf(sqrtf(s2), EPS);
    if (m == r) { inv1 = i1; inv2 = i2; }
  }

  const float* A2 = g2  + (size_t)(row0 + m) * D;
  const float* A1 = g1  + (size_t)(row0 + m) * D;
  const float* B1 = hs1 + (size_t)(row0 + m) * D;   // n == m for this lane
  const float* B2 = hs2 + (size_t)(row0 + m) * D;

  v8f c1 = {0,0,0,0,0,0,0,0};
  v8f c2 = {0,0,0,0,0,0,0,0};

#pragma unroll 4
  for (int kk = 0; kk < D; kk += 4) {
    const int ko = kk + 2 * hv;
    v2f a2 = *(const v2f*)(A2 + ko); a2 *= inv2;
    v2f b1 = *(const v2f*)(B1 + ko);
    c2 = __builtin_amdgcn_wmma_f32_16x16x4_f32(false, a2, false, b1,
                                               (short)0, c2, false, false);
    v2f a1 = *(const v2f*)(A1 + ko); a1 *= inv1;
    v2f b2 = *(const v2f*)(B2 + ko);
    c1 = __builtin_amdgcn_wmma_f32_16x16x4_f32(false, a1, false, b2,
                                               (short)0, c1, false, false);
  }

  // Diagonal of 16x16 C: (VGPR r, lane r) -> (M=r,N=r); (VGPR r, lane 24+r) -> (M=8+r,N=8+r)
  float d = 0.0f;
#pragma unroll
  for (int r = 0; r < 8; ++r) {
    float dv = c1[r] + c2[r];
    d += ((lane == r) || (lane == 24 + r)) ? dv : 0.0f;
  }
  d = wave_sum(d);
  if (lane == 0) partial[wave] = d;
}

// ---------------------------------------------------------------------------
// Kernel C: deterministic single-block tree reduction of tile partials.
// ---------------------------------------------------------------------------
__global__ __launch_bounds__(512)
void final_reduce_kernel(const float* __restrict__ partial,
                         float* __restrict__ out, int M, float scale)
{
  __shared__ float s[512];
  const int t = threadIdx.x;
  s[t] = (t < M) ? partial[t] : 0.0f;
  __syncthreads();
  for (int off = 256; off > 0; off >>= 1) {
    if (t < off) s[t] += s[t + off];
    __syncthreads();
  }
  if (t == 0) out[0] = s[0] * scale;
}

extern "C" void kernel_launch(void* const* d_in, const int* in_sizes, int n_in,
                              void* d_out, int out_size, void* d_ws, size_t ws_size,
                              hipStream_t stream)
{
  (void)n_in; (void)out_size; (void)ws_size;
  const float* h1  = (const float*)d_in[0];
  const float* h2  = (const float*)d_in[1];
  const float* g1  = (const float*)d_in[2];
  const float* g2  = (const float*)d_in[3];
  const int*   b32 = (const int*)d_in[4];

  const int N = in_sizes[0] / D;   // 500000
  const int G = in_sizes[2] / D;   // 8192

  float* hsum1   = (float*)d_ws;                     // [G, D]
  float* hsum2   = hsum1 + (size_t)G * D;            // [G, D]
  float* partial = hsum2 + (size_t)G * D;            // [G/16]

  // Kernel A: one wave per graph, 8 waves per 256-thread block.
  dim3 blkA(256), grdA((G + 7) / 8);
  seg_norm_sum_kernel<<<grdA, blkA, 0, stream>>>(h1, h2, b32, hsum1, hsum2, N, G);

  // Kernel B: one wave per 16-row tile.
  const int tiles = G / 16;
  dim3 blkB(256), grdB((tiles + 7) / 8);
  wmma_dot_kernel<<<grdB, blkB, 0, stream>>>(g1, g2, hsum1, hsum2, partial, tiles);

  // Kernel C: reduce and scale by 1/G.
  final_reduce_kernel<<<1, 512, 0, stream>>>(partial, (float*)d_out, tiles, 1.0f / (float)G);
}